// LSTMModel_77309411754
// MI455X (gfx1250) — compile-verified
//
#include <hip/hip_runtime.h>

typedef __attribute__((ext_vector_type(2))) float v2f;
typedef __attribute__((ext_vector_type(8))) float v8f;

#define TSTEPS 1024
#define HID    50
#define KPAD   52      // K padded: 13 * 4
#define NPAD   64      // per-gate padded output columns
#define ROWS   16      // batch rows per workgroup
#define NT     4       // N tiles per gate (64/16)
#define NK     13      // K steps (52/4)
#define XCHUNK 128     // x timesteps staged in LDS at a time

__global__ __launch_bounds__(128, 1) void lstm_wmma_kernel(
    const float* __restrict__ x,
    const float* __restrict__ W_ih,
    const float* __restrict__ W_hh,
    const float* __restrict__ b_ih,
    const float* __restrict__ b_hh,
    const float* __restrict__ W_fc,
    const float* __restrict__ b_fc,
    float* __restrict__ out)
{
    __shared__ float x_s[ROWS][XCHUNK];     // 8 KB staged input chunk
    __shared__ float h_s[ROWS][KPAD];       // hidden state (padded K)
    __shared__ float c_s[ROWS][HID];        // cell state
    __shared__ float g_s[4][ROWS][NPAD];    // per-gate WMMA outputs (16 KB)
    __shared__ float wih_s[4 * HID];
    __shared__ float bias_s[4 * HID];

    const int tid  = threadIdx.x;
    const int lane = tid & 31;
    const int q    = tid >> 5;      // wave id == gate id (0=i,1=f,2=g,3=o)
    const int half = lane >> 4;     // lane-half selects K offset per ISA layout
    const int nlo  = lane & 15;
    const int b0   = blockIdx.x * ROWS;

    // ---- preload weights / biases, zero-init h and c ----
    for (int i = tid; i < 4 * HID; i += 128) {
        wih_s[i]  = W_ih[i];
        bias_s[i] = b_ih[i] + b_hh[i];
    }
    for (int i = tid; i < ROWS * KPAD; i += 128) ((float*)h_s)[i] = 0.f;
    for (int i = tid; i < ROWS * HID;  i += 128) ((float*)c_s)[i] = 0.f;

    // ---- precompute time-invariant B fragments: W_hh^T gate block, 52x64 padded ----
    // B 4x16 fragment layout: VGPR0 = K(half*2), VGPR1 = K(half*2+1), N = lane&15
    v2f bfrag[NT][NK];
    #pragma unroll
    for (int nt = 0; nt < NT; ++nt) {
        const int col = nt * 16 + nlo;
        #pragma unroll
        for (int kk = 0; kk < NK; ++kk) {
            const int k = kk * 4 + half * 2;
            float e0 = 0.f, e1 = 0.f;
            if (col < HID) {
                const int grow = q * HID + col;        // row of W_hh (gate-major)
                if (k     < HID) e0 = W_hh[grow * HID + k];
                if (k + 1 < HID) e1 = W_hh[grow * HID + k + 1];
            }
            bfrag[nt][kk].x = e0;
            bfrag[nt][kk].y = e1;
        }
    }
    __syncthreads();

    const int arow = nlo;   // A-fragment row for this lane

    for (int t = 0; t < TSTEPS; ++t) {
        // ---- refill x chunk (uniform condition; barrier1 below publishes it) ----
        if ((t & (XCHUNK - 1)) == 0) {
            for (int i = tid; i < ROWS * XCHUNK; i += 128) {
                const int r  = i >> 7;
                const int tt = i & (XCHUNK - 1);
                x_s[r][tt] = x[(size_t)(b0 + r) * TSTEPS + t + tt];
            }
        }

        // ---- gates[16x64] (this wave's gate) = h[16x52] @ W_hh^T block, fp32 WMMA ----
        v8f acc[NT];
        #pragma unroll
        for (int nt = 0; nt < NT; ++nt)
            acc[nt] = (v8f){0.f, 0.f, 0.f, 0.f, 0.f, 0.f, 0.f, 0.f};

        #pragma unroll
        for (int kk = 0; kk < NK; ++kk) {
            const int k = kk * 4 + half * 2;
            v2f a;
            a.x = h_s[arow][k];
            a.y = h_s[arow][k + 1];
            #pragma unroll
            for (int nt = 0; nt < NT; ++nt) {
                acc[nt] = __builtin_amdgcn_wmma_f32_16x16x4_f32(
                    false, a, false, bfrag[nt][kk], (short)0, acc[nt], false, false);
            }
        }

        // ---- spill gate tile to LDS (D layout: VGPRv -> M = v + half*8, N = lane&15) ----
        #pragma unroll
        for (int nt = 0; nt < NT; ++nt) {
            #pragma unroll
            for (int v = 0; v < 8; ++v) {
                g_s[q][v + half * 8][nt * 16 + nlo] = acc[nt][v];
            }
        }
        __syncthreads();   // barrier1: gates (and x chunk) visible

        // ---- elementwise LSTM cell update over 16 rows x 50 cols ----
        for (int idx = tid; idx < ROWS * HID; idx += 128) {
            const int row = idx / HID;
            const int col = idx - row * HID;
            const float xv = x_s[row][t & (XCHUNK - 1)];
            const float gi = g_s[0][row][col] + xv * wih_s[0 * HID + col] + bias_s[0 * HID + col];
            const float gf = g_s[1][row][col] + xv * wih_s[1 * HID + col] + bias_s[1 * HID + col];
            const float gg = g_s[2][row][col] + xv * wih_s[2 * HID + col] + bias_s[2 * HID + col];
            const float go = g_s[3][row][col] + xv * wih_s[3 * HID + col] + bias_s[3 * HID + col];
            const float i_ = 1.f / (1.f + __expf(-gi));
            const float f_ = 1.f / (1.f + __expf(-gf));
            const float g_ = tanhf(gg);
            const float o_ = 1.f / (1.f + __expf(-go));
            const float cn = f_ * c_s[row][col] + i_ * g_;
            c_s[row][col] = cn;
            h_s[row][col] = o_ * tanhf(cn);
        }
        __syncthreads();   // barrier2: new h visible to next step's A loads
    }

    // ---- final FC: out[b] = h_last . W_fc + b_fc ----
    if (tid < ROWS) {
        float s = b_fc[0];
        for (int j = 0; j < HID; ++j) s += h_s[tid][j] * W_fc[j];
        out[b0 + tid] = s;
    }
}

extern "C" void kernel_launch(void* const* d_in, const int* in_sizes, int n_in,
                              void* d_out, int out_size, void* d_ws, size_t ws_size,
                              hipStream_t stream) {
    const float* x    = (const float*)d_in[0];
    const float* W_ih = (const float*)d_in[1];
    const float* W_hh = (const float*)d_in[2];
    const float* b_ih = (const float*)d_in[3];
    const float* b_hh = (const float*)d_in[4];
    const float* W_fc = (const float*)d_in[5];
    const float* b_fc = (const float*)d_in[6];
    float* out = (float*)d_out;

    dim3 grid(2048 / ROWS);   // 128 workgroups
    dim3 block(128);          // 4 waves: one per LSTM gate
    hipLaunchKernelGGL(lstm_wmma_kernel, grid, block, 0, stream,
                       x, W_ih, W_hh, b_ih, b_hh, W_fc, b_fc, out);
}